// UMGAD_44959717654593
// MI455X (gfx1250) — compile-verified
//
#include <hip/hip_runtime.h>
#include <hip/hip_bf16.h>
#include <math.h>

// ---------------------------------------------------------------------------
// UMGAD fused pipeline for MI455X (gfx1250, wave32)
//   Stage A: per-relation GCN norm + 2x propagation (L2-resident scatter/gather)
//   Stage B: fold softmax(a), W_enc, W_dec into one Wbig[256,512] + bias[512]
//   Stage C: single f32 WMMA GEMM  [N,256] @ [256,512] -> out [4,N,128]
// ---------------------------------------------------------------------------

typedef __attribute__((ext_vector_type(2))) float v2f;
typedef __attribute__((ext_vector_type(8))) float v8f;

#define DD 128          // feature dim (== hid dim)
#define RR 2            // relations
#define KDIM 256        // R*D  (GEMM K)
#define CDIM 512        // 4*D  (GEMM N: 3 decoders + fused[1])

// ---------------- small prep kernels ----------------------------------------

__global__ void softmax2_kernel(const float* __restrict__ a, float* __restrict__ wsoft) {
    if (threadIdx.x == 0 && blockIdx.x == 0) {
        float m  = fmaxf(a[0], a[1]);
        float e0 = expf(a[0] - m), e1 = expf(a[1] - m);
        float s  = e0 + e1;
        wsoft[0] = e0 / s;
        wsoft[1] = e1 / s;
    }
}

// Wbig[k][c]:  k = r*128+dk (over hp features), c = j*128+d (over outputs)
//  j<3 : wsoft[r] * sum_h W_enc[sel_j, r, dk, h] * W_dec[j, h, d]
//  j==3: wsoft[r] * W_enc[1, r, dk, d]
__global__ __launch_bounds__(256) void build_wbig_kernel(
        const float* __restrict__ W_enc, const float* __restrict__ W_dec,
        const float* __restrict__ wsoft, float* __restrict__ Wbig) {
    int idx = blockIdx.x * blockDim.x + threadIdx.x;
    if (idx >= KDIM * CDIM) return;
    int k = idx >> 9;          // 0..255
    int c = idx & (CDIM - 1);  // 0..511
    int r = k >> 7, dk = k & 127;
    int j = c >> 7, d  = c & 127;
    float ws = wsoft[r];
    float v;
    if (j < 3) {
        const int sel[3] = {0, 2, 3};
        int kk = sel[j];
        const float* we = W_enc + ((size_t)(kk * RR + r) * DD + dk) * DD; // contiguous over h
        const float* wd = W_dec + (size_t)j * DD * DD + d;                // stride DD over h
        float s = 0.f;
        #pragma unroll 8
        for (int h = 0; h < DD; ++h) s = fmaf(we[h], wd[(size_t)h * DD], s);
        v = ws * s;
    } else {
        v = ws * W_enc[((size_t)(1 * RR + r) * DD + dk) * DD + d];
    }
    Wbig[(size_t)k * CDIM + c] = v;
}

__global__ void build_bias_kernel(
        const float* __restrict__ b_enc, const float* __restrict__ b_dec,
        const float* __restrict__ W_dec, const float* __restrict__ wsoft,
        float* __restrict__ biasbig) {
    int c = blockIdx.x * blockDim.x + threadIdx.x;
    if (c >= CDIM) return;
    int j = c >> 7, d = c & 127;
    float v;
    if (j < 3) {
        const int sel[3] = {0, 2, 3};
        int kk = sel[j];
        float s = 0.f;
        for (int h = 0; h < DD; ++h) {
            float bf = wsoft[0] * b_enc[(kk * RR + 0) * DD + h]
                     + wsoft[1] * b_enc[(kk * RR + 1) * DD + h];
            s = fmaf(bf, W_dec[(size_t)(j * DD + h) * DD + d], s);
        }
        v = s + b_dec[j * DD + d];
    } else {
        v = wsoft[0] * b_enc[(1 * RR + 0) * DD + d]
          + wsoft[1] * b_enc[(1 * RR + 1) * DD + d];
    }
    biasbig[c] = v;
}

// ---------------- GCN normalization -----------------------------------------

__global__ void deg_init_kernel(float* __restrict__ deg, int n) {
    int i = blockIdx.x * blockDim.x + threadIdx.x;
    if (i < n) deg[i] = 1.0f;   // self loop contribution
}

__global__ void deg_accum_kernel(const int* __restrict__ col, float* __restrict__ deg, int E) {
    int e = blockIdx.x * blockDim.x + threadIdx.x;
    if (e < E) atomicAdd(&deg[col[e]], 1.0f);
}

__global__ void dinv_kernel(float* __restrict__ deg, int n) {
    int i = blockIdx.x * blockDim.x + threadIdx.x;
    if (i < n) deg[i] = rsqrtf(deg[i]);   // deg >= 1 always (self loop)
}

__global__ void edge_w_kernel(const int* __restrict__ row, const int* __restrict__ col,
                              const float* __restrict__ dinv, float* __restrict__ w, int E) {
    int e = blockIdx.x * blockDim.x + threadIdx.x;
    if (e < E) w[e] = dinv[row[e]] * dinv[col[e]];
}

// ---------------- propagation: out[row] += in[col] * w ----------------------
// one wave (32 lanes) per edge; each lane moves float4 -> 512B coalesced row

__global__ __launch_bounds__(256) void prop_kernel(
        const float* __restrict__ in, float* __restrict__ out,
        const int* __restrict__ rows, const int* __restrict__ cols,
        const float* __restrict__ w, int E, int out_stride, int out_off) {
    int wave = threadIdx.x >> 5;
    int lane = threadIdx.x & 31;
    int e = blockIdx.x * 8 + wave;
    if (e >= E) return;
    int r = rows[e], c = cols[e];
    float ww = w[e];
    const float4* src = (const float4*)(in + (size_t)c * DD) + lane;
    float4 v = *src;
    float* dst = out + (size_t)r * out_stride + out_off + lane * 4;
    atomicAdd(dst + 0, v.x * ww);
    atomicAdd(dst + 1, v.y * ww);
    atomicAdd(dst + 2, v.z * ww);
    atomicAdd(dst + 3, v.w * ww);
}

// self loop: out[i] += in[i] * dinv[i]^2   (no conflicts -> plain RMW)
__global__ __launch_bounds__(256) void selfprop_kernel(
        const float* __restrict__ in, float* __restrict__ out,
        const float* __restrict__ dinv, int n, int out_stride, int out_off) {
    int idx = blockIdx.x * blockDim.x + threadIdx.x;
    if (idx >= n * DD) return;
    int nd = idx >> 7, d = idx & 127;
    float s = dinv[nd];
    out[(size_t)nd * out_stride + out_off + d] += in[idx] * s * s;
}

// ---------------- main GEMM: f32 WMMA 16x16x4 -------------------------------
// A = hpcat [N x 256], B = Wbig [256 x 512], D written permuted into out[4,N,128].
// block = 256 threads = 8 waves; block owns 16 rows; each wave owns 4 col tiles.

__global__ __launch_bounds__(256) void gemm_wmma_kernel(
        const float* __restrict__ A, const float* __restrict__ B,
        const float* __restrict__ bias, float* __restrict__ out, int Ntot) {
    int wave  = threadIdx.x >> 5;
    int lane  = threadIdx.x & 31;
    int m     = lane & 15;            // A row / D col within tile
    int khalf = (lane >> 4) << 1;     // 0 (lanes 0-15) or 2 (lanes 16-31)
    int rt    = blockIdx.x;

    int rowA = rt * 16 + m;
    if (rowA >= Ntot) rowA = Ntot - 1;           // tail clamp (reads only)
    const float* Arow = A + (size_t)rowA * KDIM + khalf;

    int cbase = wave * 64 + m;                   // this wave: cols [wave*64, +64)
    v8f acc[4];
    #pragma unroll
    for (int t = 0; t < 4; ++t) {                // init C with bias
        float bv = bias[cbase + t * 16];
        #pragma unroll
        for (int i = 0; i < 8; ++i) acc[t][i] = bv;
    }

    for (int kb = 0; kb < KDIM; kb += 4) {
        // A frag: lane holds A[m][kb+khalf], A[m][kb+khalf+1]
        v2f a;
        a.x = Arow[kb];
        a.y = Arow[kb + 1];
        const float* Bp = B + (size_t)(kb + khalf) * CDIM;
        #pragma unroll
        for (int t = 0; t < 4; ++t) {
            int c = cbase + t * 16;
            v2f b;
            b.x = Bp[c];
            b.y = Bp[CDIM + c];
            acc[t] = __builtin_amdgcn_wmma_f32_16x16x4_f32(
                false, a, false, b, (short)0, acc[t], false, false);
        }
    }

    // D layout: VGPR i -> M = i + 8*(lane>=16), N = lane&15
    int rowbase = rt * 16 + ((lane >> 4) << 3);
    if (rt * 16 + 16 <= Ntot) {
        // full tile: unguarded, clause-friendly stores (64B contig per half-wave)
        #pragma unroll
        for (int t = 0; t < 4; ++t) {
            int c = cbase + t * 16;
            int j = c >> 7, d = c & 127;
            float* op = out + (size_t)j * Ntot * DD + (size_t)rowbase * DD + d;
            #pragma unroll
            for (int i = 0; i < 8; ++i) {
                op[(size_t)i * DD] = acc[t][i];
            }
        }
    } else {
        // tail tile (only if N % 16 != 0)
        #pragma unroll
        for (int t = 0; t < 4; ++t) {
            int c = cbase + t * 16;
            int j = c >> 7, d = c & 127;
            float* op = out + (size_t)j * Ntot * DD + d;
            #pragma unroll
            for (int i = 0; i < 8; ++i) {
                int row = rowbase + i;
                if (row < Ntot) op[(size_t)row * DD] = acc[t][i];
            }
        }
    }
}

// ---------------------------------------------------------------------------

extern "C" void kernel_launch(void* const* d_in, const int* in_sizes, int n_in,
                              void* d_out, int out_size, void* d_ws, size_t ws_size,
                              hipStream_t stream) {
    const float* x     = (const float*)d_in[0];   // [N,128]
    const int*   ei    = (const int*)d_in[1];     // [R,2,E]
    const float* W_enc = (const float*)d_in[2];   // [4,R,128,128]
    const float* b_enc = (const float*)d_in[3];   // [4,R,128]
    const float* W_dec = (const float*)d_in[4];   // [3,128,128]
    const float* b_dec = (const float*)d_in[5];   // [3,128]
    const float* a     = (const float*)d_in[6];   // [R]
    float* out = (float*)d_out;                   // [4,N,128]

    const int N = in_sizes[0] / DD;
    const int E = in_sizes[1] / (2 * RR);

    // workspace layout (floats)
    float* hpcat   = (float*)d_ws;                      // [N,256]
    float* h1      = hpcat + (size_t)N * KDIM;          // [N,128]
    float* dinv    = h1 + (size_t)N * DD;               // [N] (deg then dinv)
    float* wbuf    = dinv + N;                          // [E]
    float* Wbig    = wbuf + E;                          // [256,512]
    float* biasbig = Wbig + (size_t)KDIM * CDIM;        // [512]
    float* wsoft   = biasbig + CDIM;                    // [2]

    // Stage B prep (independent of propagation; runs in stream order)
    hipMemsetAsync(hpcat, 0, (size_t)N * KDIM * sizeof(float), stream);
    softmax2_kernel<<<1, 32, 0, stream>>>(a, wsoft);
    build_wbig_kernel<<<(KDIM * CDIM + 255) / 256, 256, 0, stream>>>(W_enc, W_dec, wsoft, Wbig);
    build_bias_kernel<<<(CDIM + 255) / 256, 256, 0, stream>>>(b_enc, b_dec, W_dec, wsoft, biasbig);

    const int nBlkN  = (N + 255) / 256;
    const int nBlkE  = (E + 255) / 256;
    const int nBlkP  = (E + 7) / 8;          // 8 edges (waves) per 256-thread block
    const int nBlkND = (int)(((size_t)N * DD + 255) / 256);

    // Stage A: per-relation GCN norm + 2 propagations
    for (int r = 0; r < RR; ++r) {
        const int* rows = ei + (size_t)r * 2 * E;
        const int* cols = rows + E;

        deg_init_kernel<<<nBlkN, 256, 0, stream>>>(dinv, N);
        deg_accum_kernel<<<nBlkE, 256, 0, stream>>>(cols, dinv, E);
        dinv_kernel<<<nBlkN, 256, 0, stream>>>(dinv, N);
        edge_w_kernel<<<nBlkE, 256, 0, stream>>>(rows, cols, dinv, wbuf, E);

        // hop 1: x -> h1
        hipMemsetAsync(h1, 0, (size_t)N * DD * sizeof(float), stream);
        prop_kernel<<<nBlkP, 256, 0, stream>>>(x, h1, rows, cols, wbuf, E, DD, 0);
        selfprop_kernel<<<nBlkND, 256, 0, stream>>>(x, h1, dinv, N, DD, 0);

        // hop 2: h1 -> hpcat[:, r*128 : r*128+128]
        prop_kernel<<<nBlkP, 256, 0, stream>>>(h1, hpcat, rows, cols, wbuf, E, KDIM, r * DD);
        selfprop_kernel<<<nBlkND, 256, 0, stream>>>(h1, hpcat, dinv, N, KDIM, r * DD);
    }

    // Stage C: fused encoder+fusion+decoder GEMM via f32 WMMA
    gemm_wmma_kernel<<<(N + 15) / 16, 256, 0, stream>>>(hpcat, Wbig, biasbig, out, N);
}